// Conv2d_with_GivenKernel_21646635171979
// MI455X (gfx1250) — compile-verified
//
#include <hip/hip_runtime.h>

// Grouped conv [16,1,512,512] * [16,32,9,9] (pad 3) -> [16,32,510,510], all f32.
// Per-batch GEMM on V_WMMA_F32_16X16X4_F32: M=32 (j), K=81 taps, N=510*510 px.
//
// Interior kernel: one wave32 computes FOUR vertically-adjacent 16x16 D tiles.
// K-order per tile: 9 rows x 2 K4 windows (kx {0..3},{4..7}) + 3 trailing
// K4 chunks covering the kx=8 column (K slot t = 4kc+2h+v = tap row), i.e.
// the optimal 21 WMMA per tile. All loads are base_ptr[compile-time-const];
// A fragments preloaded once and shared by all 4 accumulators; the 12 shared
// input rows amortize B traffic 3x vs. one-tile waves.

typedef __attribute__((ext_vector_type(2))) float v2f;
typedef __attribute__((ext_vector_type(8))) float v8f;

#define BATCH 16
#define JCH   32
#define KWID  9
#define KTAP  81
#define H     512
#define W     512
#define OH    510
#define OW    510
#define PAD   3

// Interior: oy in [3,506] (504 rows = 126 groups of 4), seg in [1,30].
#define ISEG  30
#define IGRP  126
// Edge tiles per (b,mt): 6 rows x 32 segs + 504 rows x 2 segs = 1200.
#define ETASKS_PER_BM 1200
#define ETOTAL (BATCH * 2 * ETASKS_PER_BM)        // 38400 waves
#define EWAVES_PER_BLOCK 8

#define WMMA_F32X4(Cacc, Av, Bv)                                            \
  __builtin_amdgcn_wmma_f32_16x16x4_f32(false, (Av), false, (Bv),           \
                                        (short)0, (Cacc), false, false)

// ---------------------------------------------------------------------------
// Interior kernel: grid (30 segs, 63 grp-pairs, 32 b*mt), 2 waves per block.
// ---------------------------------------------------------------------------
__global__ __launch_bounds__(64)
void conv_wmma_interior(const float* __restrict__ in,
                        const float* __restrict__ wgt,
                        float* __restrict__ out) {
  const int lane = threadIdx.x & 31;
  const int wave = threadIdx.x >> 5;          // 0..1
  const int l    = lane & 15;
  const int half = lane >> 4;

  const int seg = 1 + blockIdx.x;             // 1..30
  const int grp = (blockIdx.y << 1) + wave;   // 0..125
  const int bm  = blockIdx.z;                 // 0..31
  const int b   = bm >> 1;
  const int mt  = bm & 1;
  const int oy0 = 3 + (grp << 2);             // 3,7,...,503
  const int oxb = seg << 4;

  const float* inb  = in  + (size_t)b * H * W;
  const float* wrow = wgt + ((size_t)b * JCH + (size_t)mt * 16 + l) * KTAP;
  const float* wptr = wrow + 2 * half;
  // Main-window B base: row iy = oy0-3+r, column oxb-3 + l + (4s + 2h + v)
  const float* bp   = inb + (oy0 - 3) * W + (oxb - 3 + l + 2 * half);
  // Trailing (kx=8) B bases: slot t=4kc+2h+v -> row oy0+a-3+t, col oxb+5+l
  const float* bc   = inb + (oy0 - 3 + 2 * half) * W + (oxb + 5 + l); // kc=0,1
  const float* bc2  = inb + (oy0 + 5) * W            + (oxb + 5 + l); // kc=2 (t=8)
  const float* w2   = wrow + 18 * half + 8;

  // ---- Preload A fragments (unconditional loads; zero-slots via multiply).
  v2f Aw[9][2];
  #pragma unroll
  for (int ky = 0; ky < 9; ++ky) {
    Aw[ky][0] = v2f{ wptr[9 * ky],     wptr[9 * ky + 1] };
    Aw[ky][1] = v2f{ wptr[9 * ky + 4], wptr[9 * ky + 5] };
  }
  const float wsel = half ? 0.0f : 1.0f;       // kc=2 chunk: only (h0,v0) real
  v2f At0 = v2f{ w2[0],  w2[9]  };             // taps (t=2h,   8), (t=2h+1, 8)
  v2f At1 = v2f{ w2[36], w2[45] };             // taps (t=4+2h, 8), (t=5+2h, 8)
  v2f At2 = v2f{ wsel * wrow[80], 0.0f };      // tap  (t=8, 8) on h=0,v=0 only

  v8f acc[4];
  #pragma unroll
  for (int a = 0; a < 4; ++a) acc[a] = v8f{0.f,0.f,0.f,0.f,0.f,0.f,0.f,0.f};

  // ---- Main windows: 12 shared rows; row r feeds acc[a] via ky = r - a.
  #pragma unroll
  for (int r = 0; r < 12; ++r) {
    v2f B0 = v2f{ bp[r * W],     bp[r * W + 1] };
    v2f B1 = v2f{ bp[r * W + 4], bp[r * W + 5] };
    #pragma unroll
    for (int a = 0; a < 4; ++a) {            // chain distance 4 per acc
      const int ky = r - a;
      if (ky >= 0 && ky < 9) acc[a] = WMMA_F32X4(acc[a], Aw[ky][0], B0);
    }
    #pragma unroll
    for (int a = 0; a < 4; ++a) {
      const int ky = r - a;
      if (ky >= 0 && ky < 9) acc[a] = WMMA_F32X4(acc[a], Aw[ky][1], B1);
    }
  }

  // ---- Trailing kx=8 column: chunk (a,kc) uses B rows x=a+4kc, x+1 (CSE'd).
  #pragma unroll
  for (int kc = 0; kc < 2; ++kc)
    #pragma unroll
    for (int a = 0; a < 4; ++a) {
      const int x = a + 4 * kc;
      v2f Bv = v2f{ bc[x * W], bc[x * W + W] };
      acc[a] = WMMA_F32X4(acc[a], kc == 0 ? At0 : At1, Bv);
    }
  #pragma unroll
  for (int a = 0; a < 4; ++a) {              // kc=2: only t=8 slot is real
    float b0 = bc2[a * W];
    v2f Bv = v2f{ b0, b0 };
    acc[a] = WMMA_F32X4(acc[a], At2, Bv);
  }

  // ---- Store 4 rows x 16 cols x 16 j's. seg<=30 -> oxb+15 <= 495: no mask.
  float* op = out + (((size_t)b * JCH + (size_t)mt * 16) * OH + oy0) * OW + oxb + l;
  #pragma unroll
  for (int a = 0; a < 4; ++a)
    #pragma unroll
    for (int v = 0; v < 8; ++v)
      __builtin_nontemporal_store(acc[a][v],
                                  op + (size_t)(v + 8 * half) * OH * OW + a * OW);
}

// ---------------------------------------------------------------------------
// Edge kernel: clamped addresses + cndmask selects (EXEC all-ones at WMMAs).
// Covers oy in {0,1,2,507,508,509} (all segs) and seg in {0,31} (other rows).
// ---------------------------------------------------------------------------
__global__ __launch_bounds__(256)
void conv_wmma_edge(const float* __restrict__ in,
                    const float* __restrict__ wgt,
                    float* __restrict__ out) {
  const int lane = threadIdx.x & 31;
  const int wave = threadIdx.x >> 5;
  const int task = blockIdx.x * EWAVES_PER_BLOCK + wave;

  int b  = task / (2 * ETASKS_PER_BM);
  int r  = task - b * (2 * ETASKS_PER_BM);
  int mt = r / ETASKS_PER_BM;
  r     -= mt * ETASKS_PER_BM;
  int oy, seg;
  if (r < 192) {                       // 6 edge rows x 32 segs
    int rr = r >> 5;                   // 0..5
    seg = r & 31;
    oy  = (rr < 3) ? rr : (504 + rr);  // {0,1,2,507,508,509}
  } else {                             // border segs of interior rows
    int e = r - 192;
    oy  = 3 + (e >> 1);
    seg = (e & 1) ? 31 : 0;
  }
  int oxb = seg * 16;

  const int l    = lane & 15;
  const int half = lane >> 4;

  const float* inb  = in  + (size_t)b * H * W;
  const float* wrow = wgt + ((size_t)b * JCH + (size_t)mt * 16 + l) * KTAP;

  v8f c = {0.f, 0.f, 0.f, 0.f, 0.f, 0.f, 0.f, 0.f};

  #pragma unroll
  for (int kc = 0; kc < 21; ++kc) {
    const int kb = kc * 4 + half * 2;
    float av[2], bv[2];
    #pragma unroll
    for (int v = 0; v < 2; ++v) {
      const int k       = kb + v;
      const int valid_k = (k < KTAP) ? 1 : 0;
      const int kk      = valid_k ? k : 0;
      const int ky      = kk / KWID;
      const int kx      = kk - ky * KWID;

      float a = wrow[kk];
      av[v] = valid_k ? a : 0.0f;

      int iy = oy  + ky - PAD;
      int ix = oxb + l + kx - PAD;
      int ok = (iy >= 0) & (iy < H) & (ix >= 0) & (ix < W) & valid_k;
      int iyc = iy < 0 ? 0 : (iy > H - 1 ? H - 1 : iy);
      int ixc = ix < 0 ? 0 : (ix > W - 1 ? W - 1 : ix);
      float x = inb[iyc * W + ixc];
      bv[v] = ok ? x : 0.0f;
    }
    v2f A  = {av[0], av[1]};
    v2f Bv = {bv[0], bv[1]};
    c = WMMA_F32X4(c, A, Bv);
  }

  if (oxb + l < OW) {
    float* op = out + (((size_t)b * JCH + (size_t)mt * 16) * OH + oy) * OW + oxb + l;
    #pragma unroll
    for (int v = 0; v < 8; ++v)
      __builtin_nontemporal_store(c[v], op + (size_t)(v + 8 * half) * OH * OW);
  }
}

extern "C" void kernel_launch(void* const* d_in, const int* in_sizes, int n_in,
                              void* d_out, int out_size, void* d_ws, size_t ws_size,
                              hipStream_t stream) {
  (void)in_sizes; (void)n_in; (void)out_size; (void)d_ws; (void)ws_size;
  const float* in  = (const float*)d_in[0];   // [16, 1, 512, 512] f32
  const float* wgt = (const float*)d_in[1];   // [16, 32, 9, 9]    f32
  float*       out = (float*)d_out;           // [16, 32, 510, 510] f32

  // Interior: 30 segs x 126 row-groups x 32 (b,mt); 2 waves per 64-thr block.
  conv_wmma_interior<<<dim3(ISEG, IGRP / 2, BATCH * 2), dim3(64), 0, stream>>>(in, wgt, out);
  // Edge: 38400 waves, 8 per 256-thr block.
  conv_wmma_edge<<<dim3(ETOTAL / EWAVES_PER_BLOCK), dim3(256), 0, stream>>>(in, wgt, out);
}